// LPFMA_48945447305224
// MI455X (gfx1250) — compile-verified
//
#include <hip/hip_runtime.h>

typedef float v2f __attribute__((ext_vector_type(2)));
typedef float v8f __attribute__((ext_vector_type(8)));

#define B_  8
#define C_  64
#define N_  24576
#define K_  16
#define BN_ (B_ * N_)

// ---------------- kernel 0: zero the BN accumulators (128 floats) ------------
__global__ void k0_zero(float* __restrict__ acc) {
    acc[threadIdx.x] = 0.0f;
}

// ---------------- kernel 1: pointwise GEMMs via V_WMMA_F32_16X16X4_F32 -------
// f: [B,64,N] channel-major.  Writes f1/f2/f3 channels-last: [B*N][64].
// One wave handles one (b, 16-point tile); computes 4 o-tiles x 3 matrices.
__global__ __launch_bounds__(256) void k1_gemm(
    const float* __restrict__ f,
    const float* __restrict__ W1, const float* __restrict__ b1,
    const float* __restrict__ W2, const float* __restrict__ b2,
    const float* __restrict__ W3, const float* __restrict__ b3,
    float* __restrict__ f1ws, float* __restrict__ f2ws, float* __restrict__ f3ws)
{
    const int lane = threadIdx.x & 31;
    const int wid  = threadIdx.x >> 5;
    const int task = blockIdx.x * 8 + wid;       // 12288 tasks total
    const int b    = task / (N_ / 16);
    const int n0   = (task % (N_ / 16)) * 16;
    const int hi   = lane >> 4;                  // which half-wave
    const int lm   = lane & 15;

    // B-fragments: f tile [64 x 16] as 16 chunks of K=4.
    // Layout: vgpr v of chunk k holds element (K = 4k + v + 2*hi, N = lm).
    v2f bf[16];
    const float* fb = f + (size_t)b * C_ * N_;
    #pragma unroll
    for (int k = 0; k < 16; ++k) {
        const int row = 4 * k + 2 * hi;
        bf[k].x = fb[(size_t)row       * N_ + n0 + lm];
        bf[k].y = fb[(size_t)(row + 1) * N_ + n0 + lm];
    }

    const float* Ws[3]  = {W1, W2, W3};
    const float* bs[3]  = {b1, b2, b3};
    float*       dst[3] = {f1ws, f2ws, f3ws};

    #pragma unroll
    for (int m = 0; m < 3; ++m) {
        const float* W    = Ws[m];
        const float* bias = bs[m];
        #pragma unroll
        for (int ot = 0; ot < 4; ++ot) {
            const int o0 = ot * 16;
            // C init = bias: acc[r] is element (M = r + 8*hi, N = lm)
            v8f acc;
            #pragma unroll
            for (int r = 0; r < 8; ++r) acc[r] = bias[o0 + 8 * hi + r];
            #pragma unroll
            for (int k = 0; k < 16; ++k) {
                // A fragment: lane holds (M = lm, K = 4k + v + 2*hi), W row-major
                v2f a = *(const v2f*)(W + (size_t)(o0 + lm) * 64 + 4 * k + 2 * hi);
                acc = __builtin_amdgcn_wmma_f32_16x16x4_f32(
                        false, a, false, bf[k], (short)0, acc, false, false);
            }
            // D lane-row r -> channel o0 + r + 8*hi at point n0+lm: 8 contiguous floats
            float* p = dst[m] + ((size_t)b * N_ + n0 + lm) * 64 + o0 + 8 * hi;
            *(float4*)(p)     = make_float4(acc[0], acc[1], acc[2], acc[3]);
            *(float4*)(p + 4) = make_float4(acc[4], acc[5], acc[6], acc[7]);
        }
    }
}

// ---------------- kernel 2: neighbor max-gather + BN partial reduction -------
// x = (max_k f1[idx] - f1) + max_k f2[idx] + f3 ; written in place over f3ws.
// One wave per point-chunk of 16; lane owns 2 channels (float2).
__global__ __launch_bounds__(256) void k2_edge(
    const int* __restrict__ idx,
    const float* __restrict__ f1ws, const float* __restrict__ f2ws,
    float* __restrict__ f3ws,
    float* __restrict__ gsum, float* __restrict__ gsq)
{
    __shared__ float sS[64];
    __shared__ float sQ[64];
    const int lane = threadIdx.x & 31;
    const int wid  = threadIdx.x >> 5;
    if (threadIdx.x < 64) { sS[threadIdx.x] = 0.0f; sQ[threadIdx.x] = 0.0f; }
    __syncthreads();

    float2 s = make_float2(0.f, 0.f), q = make_float2(0.f, 0.f);
    const int base_pt = blockIdx.x * 128 + wid * 16;

    #pragma unroll 1
    for (int p = 0; p < 16; ++p) {
        const int    pt    = base_pt + p;
        const size_t bbase = (size_t)(pt / N_) * N_;
        const float2 f1v = ((const float2*)f1ws)[(size_t)pt * 32 + lane];
        const float2 f3v = ((const float2*)f3ws)[(size_t)pt * 32 + lane];
        const float  NI  = -__builtin_inff();
        float2 m1 = make_float2(NI, NI), m2 = make_float2(NI, NI);
        const int* ip = idx + (size_t)pt * 16;
        #pragma unroll
        for (int k = 0; k < 16; ++k) {
            const size_t j = bbase + (size_t)ip[k];
            const float2 g1 = ((const float2*)f1ws)[j * 32 + lane];
            const float2 g2 = ((const float2*)f2ws)[j * 32 + lane];
            m1.x = fmaxf(m1.x, g1.x); m1.y = fmaxf(m1.y, g1.y);
            m2.x = fmaxf(m2.x, g2.x); m2.y = fmaxf(m2.y, g2.y);
        }
        float2 x;
        x.x = (m1.x - f1v.x) + m2.x + f3v.x;
        x.y = (m1.y - f1v.y) + m2.y + f3v.y;
        ((float2*)f3ws)[(size_t)pt * 32 + lane] = x;
        s.x += x.x;        s.y += x.y;
        q.x += x.x * x.x;  q.y += x.y * x.y;
    }

    atomicAdd(&sS[2 * lane],     s.x);
    atomicAdd(&sS[2 * lane + 1], s.y);
    atomicAdd(&sQ[2 * lane],     q.x);
    atomicAdd(&sQ[2 * lane + 1], q.y);
    __syncthreads();
    if (threadIdx.x < 64) {
        atomicAdd(&gsum[threadIdx.x], sS[threadIdx.x]);
        atomicAdd(&gsq[threadIdx.x],  sQ[threadIdx.x]);
    }
}

// ---------------- kernel 3: BN normalize + layout transpose ------------------
// Reads x channels-last, LDS-transposes a 64x64 tile, writes out [B,C,N].
__global__ __launch_bounds__(256) void k3_bn(
    const float* __restrict__ xws,
    const float* __restrict__ gsum, const float* __restrict__ gsq,
    const float* __restrict__ gamma, const float* __restrict__ beta,
    float* __restrict__ out)
{
    __shared__ float tile[64 * 65];
    const int blocksPerB = N_ / 64;                    // 384
    const int b  = blockIdx.x / blocksPerB;
    const int n0 = (blockIdx.x % blocksPerB) * 64;
    const float* src = xws + ((size_t)b * N_ + n0) * 64;

    #pragma unroll
    for (int r = 0; r < 16; ++r) {
        const int e = threadIdx.x + 256 * r;
        const int i = e >> 6, c = e & 63;
        tile[i * 65 + c] = src[e];                     // coalesced channels-last read
    }
    __syncthreads();

    const float invCnt = 1.0f / (float)BN_;
    #pragma unroll
    for (int r = 0; r < 16; ++r) {
        const int e = threadIdx.x + 256 * r;
        const int c = e >> 6, i = e & 63;
        const float mean = gsum[c] * invCnt;
        const float var  = gsq[c] * invCnt - mean * mean;
        const float xv   = tile[i * 65 + c];
        const float yv   = (xv - mean) * rsqrtf(var + 1e-5f) * gamma[c] + beta[c];
        out[((size_t)b * 64 + c) * N_ + n0 + i] = yv;  // coalesced along n
    }
}

// -----------------------------------------------------------------------------
extern "C" void kernel_launch(void* const* d_in, const int* in_sizes, int n_in,
                              void* d_out, int out_size, void* d_ws, size_t ws_size,
                              hipStream_t stream) {
    const float* f     = (const float*)d_in[0];
    const int*   idx   = (const int*)  d_in[1];
    const float* W1    = (const float*)d_in[2];
    const float* b1    = (const float*)d_in[3];
    const float* W2    = (const float*)d_in[4];
    const float* b2    = (const float*)d_in[5];
    const float* W3    = (const float*)d_in[6];
    const float* b3    = (const float*)d_in[7];
    const float* gamma = (const float*)d_in[8];
    const float* beta  = (const float*)d_in[9];
    float* out = (float*)d_out;

    // workspace: f1 | f2 | f3(->x in place) | sum[64] | sumsq[64]  (~151 MB)
    float* ws   = (float*)d_ws;
    const size_t SEG = (size_t)BN_ * 64;
    float* f1ws = ws;
    float* f2ws = ws + SEG;
    float* f3ws = ws + 2 * SEG;
    float* gsum = ws + 3 * SEG;
    float* gsq  = gsum + 64;

    k0_zero<<<1, 128, 0, stream>>>(gsum);
    k1_gemm<<<1536, 256, 0, stream>>>(f, W1, b1, W2, b2, W3, b3, f1ws, f2ws, f3ws);
    k2_edge<<<1536, 256, 0, stream>>>(idx, f1ws, f2ws, f3ws, gsum, gsq);
    k3_bn  <<<3072, 256, 0, stream>>>(f3ws, gsum, gsq, gamma, beta, out);
}